// AtomicBasis_85263690760900
// MI455X (gfx1250) — compile-verified
//
#include <hip/hip_runtime.h>
#include <hip/hip_bf16.h>

typedef float v2f __attribute__((ext_vector_type(2)));
typedef float v8f __attribute__((ext_vector_type(8)));

#define N_NEIGH 4096
#define N_CH    64
#define NCOMP   13          // 1 + 3 + 9 output tensor components
#define NOUT    832         // 64 + 192 + 576
#define BLOCKS  32
#define TPB     256
#define WAVES   8           // per block (wave32)
#define STREAMS (BLOCKS * 2)            // 64 neighbor streams (2 per block)
#define NTILES  (N_NEIGH / 4)           // 1024 tiles of 4 neighbors
#define TILES_PER_STREAM (NTILES / STREAMS)  // 16

__global__ __launch_bounds__(TPB) void ab_partial_kernel(
    const float* __restrict__ h0,
    const float* __restrict__ h1,
    const float* __restrict__ h2,
    const float* __restrict__ rel,
    float* __restrict__ ws)
{
  __shared__ float s_dist[N_NEIGH];                 // 16 KB: full distance table
  __shared__ float s_red[WAVES][NCOMP][16];         // per-wave partials

  // Cooperative: distances for ALL neighbors (radial term indexes them crosswise)
  for (int n = threadIdx.x; n < N_NEIGH; n += TPB) {
    float x = rel[n * 3 + 0];
    float y = rel[n * 3 + 1];
    float z = rel[n * 3 + 2];
    s_dist[n] = sqrtf(x * x + y * y + z * z);
  }
  __syncthreads();

  const int lane  = threadIdx.x & 31;
  const int wave  = threadIdx.x >> 5;
  const int cg    = wave & 3;          // channel group (16 channels each)
  const int part  = wave >> 2;         // neighbor partition within block
  const int strm  = blockIdx.x * 2 + part;
  const int c     = cg * 16 + (lane & 15);
  const int khalf = lane >> 4;         // K-pair: lanes 0-15 hold k=0,1; 16-31 hold k=2,3

  v8f acc[NCOMP] = {};                 // 13 WMMA accumulators (f32 16x16)
  v2f ones; ones[0] = 1.0f; ones[1] = 1.0f;

  for (int t = 0; t < TILES_PER_STREAM; ++t) {
    const int tile = strm + t * STREAMS;
    const int nb   = tile * 4 + khalf * 2;

    // Prefetch next tile's h2 stream (global_prefetch_b8)
    if (t + 1 < TILES_PER_STREAM) {
      const int nn = (strm + (t + 1) * STREAMS) * 4 + khalf * 2;
      __builtin_prefetch((const void*)(h2 + (size_t)(nn * N_CH + c) * 9), 0, 1);
    }

    float yv2[2][NCOMP];
#pragma unroll
    for (int kk = 0; kk < 2; ++kk) {
      const int n   = nb + kk;
      const int idx = n * N_CH + c;

      // unit direction + |u|^2 (kept explicit for faithfulness)
      const float rx = rel[n * 3 + 0], ry = rel[n * 3 + 1], rz = rel[n * 3 + 2];
      const float inv = 1.0f / s_dist[n];
      const float ux = rx * inv, uy = ry * inv, uz = rz * inv;
      const float s = ux * ux + uy * uy + uz * uz;
      const float one_s = 1.0f + s;

      // radial: faithful (C,N)->(N,C) reshape => uses dist of neighbor m, basis n>>6 + 1
      const int   m   = ((n & 63) << 6) + c;
      const float dm  = s_dist[m];
      const float nb_ = (float)((n >> 6) + 1);
      const float rad = 0.63245553203f * __sinf(nb_ * 0.62831853072f * dm) / dm;

      const float h0v = h0[idx];
      const float a0 = h1[idx * 3 + 0], a1 = h1[idx * 3 + 1], a2 = h1[idx * 3 + 2];
      const float* H2 = h2 + (size_t)idx * 9;
      const float m00 = H2[0], m01 = H2[1], m02 = H2[2];
      const float m10 = H2[3], m11 = H2[4], m12 = H2[5];
      const float m20 = H2[6], m21 = H2[7], m22 = H2[8];

      const float tr  = m00 + m11 + m22;
      const float h1u = a0 * ux + a1 * uy + a2 * uz;
      const float w0  = m00 * ux + m01 * uy + m02 * uz;   // h2 · u
      const float w1  = m10 * ux + m11 * uy + m12 * uz;
      const float w2  = m20 * ux + m21 * uy + m22 * uz;
      const float t0  = m00 * ux + m10 * uy + m20 * uz;   // h2ᵀ · u
      const float t1  = m01 * ux + m11 * uy + m21 * uz;
      const float t2  = m02 * ux + m12 * uy + m22 * uz;
      const float uh2u = ux * w0 + uy * w1 + uz * w2;

      const float u[3]  = {ux, uy, uz};
      const float hv[3] = {a0, a1, a2};
      const float w[3]  = {w0, w1, w2};
      const float wt[3] = {t0, t1, t2};
      const float h2m[3][3] = {{m00, m01, m02}, {m10, m11, m12}, {m20, m21, m22}};

      float* yv = yv2[kk];
      yv[0] = rad * (h0v * one_s + h1u + tr * one_s + 2.0f * uh2u);
#pragma unroll
      for (int d = 0; d < 3; ++d) {
        yv[1 + d] = rad * (h0v * u[d] + hv[d] * one_s + 2.0f * h1u * u[d]
                           + tr * u[d] + w[d] + wt[d]);
#pragma unroll
        for (int e = 0; e < 3; ++e) {
          yv[4 + d * 3 + e] = rad * (h0v * u[d] * u[e] + hv[d] * u[e]
                                     + h2m[d][e] * one_s + tr * u[d] * u[e]
                                     + 2.0f * (w[d] + wt[d]) * u[e]);
        }
      }
    }

    // WMMA-based reduction: D = ones(16x4) x B(4x16) + C  -> column sums per channel
#pragma unroll
    for (int j = 0; j < NCOMP; ++j) {
      v2f b; b[0] = yv2[0][j]; b[1] = yv2[1][j];
      acc[j] = __builtin_amdgcn_wmma_f32_16x16x4_f32(
          /*neg_a=*/false, ones, /*neg_b=*/false, b,
          /*c_mod=*/(short)0, acc[j], /*reuse_a=*/false, /*reuse_b=*/false);
    }
  }

  // Row 0 of D (VGPR0, lanes 0-15) holds the per-channel sums
  if (lane < 16) {
#pragma unroll
    for (int j = 0; j < NCOMP; ++j) s_red[wave][j][lane] = acc[j][0];
  }
  __syncthreads();

  // Block partials in final output layout: [0,64)=rank0, [64,256)=rank1, [256,832)=rank2
  for (int o = threadIdx.x; o < NOUT; o += TPB) {
    int cc, j;
    if (o < 64)       { cc = o;             j = 0; }
    else if (o < 256) { int r = o - 64;  cc = r / 3; j = 1 + r % 3; }
    else              { int r = o - 256; cc = r / 9; j = 4 + r % 9; }
    const int g = cc >> 4, l = cc & 15;
    ws[blockIdx.x * NOUT + o] = s_red[g][j][l] + s_red[4 + g][j][l];
  }
}

__global__ __launch_bounds__(256) void ab_reduce_kernel(
    const float* __restrict__ ws, float* __restrict__ out)
{
  for (int o = threadIdx.x; o < NOUT; o += 256) {
    float s = 0.0f;
    for (int b = 0; b < BLOCKS; ++b) s += ws[b * NOUT + o];
    out[o] = s;
  }
}

extern "C" void kernel_launch(void* const* d_in, const int* in_sizes, int n_in,
                              void* d_out, int out_size, void* d_ws, size_t ws_size,
                              hipStream_t stream) {
  const float* h0  = (const float*)d_in[0];
  const float* h1  = (const float*)d_in[1];
  const float* h2  = (const float*)d_in[2];
  const float* rel = (const float*)d_in[3];
  // d_in[4] (channel_weights, 201 MB) is dead code in the reference: never read.
  float* ws = (float*)d_ws;   // needs 32 * 832 * 4 = 106,496 bytes

  ab_partial_kernel<<<BLOCKS, TPB, 0, stream>>>(h0, h1, h2, rel, ws);
  ab_reduce_kernel<<<1, 256, 0, stream>>>(ws, (float*)d_out);
}